// DAEGCModel_66039417143761
// MI455X (gfx1250) — compile-verified
//
#include <hip/hip_runtime.h>
#include <hip/hip_bf16.h>
#include <math.h>

typedef __attribute__((ext_vector_type(2))) float v2f;
typedef __attribute__((ext_vector_type(8))) float v8f;

// ---------------- order-preserving float<->int for atomicMax ----------------
__device__ __forceinline__ int ordf(float f) {
    int i = __float_as_int(f);
    return i >= 0 ? i : (i ^ 0x7FFFFFFF);
}
__device__ __forceinline__ float unordf(int o) {
    int i = o >= 0 ? o : (o ^ 0x7FFFFFFF);
    return __int_as_float(i);
}

// ---------------- generic f32 WMMA GEMM: C[M,N] = A[M,K] * B[K,N] ----------
// one wave computes one 16x16 tile; K stepped by 4 via V_WMMA_F32_16X16X4_F32
__global__ void wmma_gemm_f32(const float* __restrict__ A, const float* __restrict__ B,
                              float* __restrict__ C, int M, int N, int Klen,
                              int lda, int ldb, int ldc)
{
    int wave   = (int)((blockIdx.x * blockDim.x + threadIdx.x) >> 5);
    int lane   = (int)(threadIdx.x & 31);
    int tilesN = N >> 4;
    int tm = wave / tilesN, tn = wave % tilesN;
    if (tm >= (M >> 4)) return;               // wave-uniform guard
    int row0 = tm << 4, col0 = tn << 4;
    int r = lane & 15, hi = lane >> 4;        // hi=0: K 0,1 ; hi=1: K 2,3
    v8f c = {};
    const float* Arow = A + (size_t)(row0 + r) * lda;
    for (int k0 = 0; k0 < Klen; k0 += 4) {
        v2f a = *(const v2f*)(Arow + k0 + 2 * hi);
        v2f b;
        b.x = B[(size_t)(k0 + 2 * hi) * ldb + col0 + r];
        b.y = B[(size_t)(k0 + 2 * hi + 1) * ldb + col0 + r];
        c = __builtin_amdgcn_wmma_f32_16x16x4_f32(false, a, false, b, (short)0, c, false, false);
    }
    #pragma unroll
    for (int v = 0; v < 8; ++v)
        C[(size_t)(row0 + v + 8 * hi) * ldc + col0 + r] = c[v];
}

// ---------------- A_pred = sigmoid(Z * Z^T), LDS macro-tiled ----------------
// Block = 256 threads (8 waves) computes a 128x64 tile of P.
// LDS panels: As[128][68], Bs[64][68] (pad 68 -> bank = 4*row + k, conflict-free)
#define ZPAD 68
__global__ __launch_bounds__(256) void wmma_zzt_sigmoid(const float* __restrict__ Z,
                                                        float* __restrict__ P,
                                                        int Nn, int D)
{
    __shared__ float As[128 * ZPAD];
    __shared__ float Bs[64 * ZPAD];

    int nbx  = Nn >> 6;                 // column blocks of 64
    int row0 = (blockIdx.x / nbx) << 7; // 128-row macro tile
    int col0 = (blockIdx.x % nbx) << 6; // 64-col macro tile
    int tid  = threadIdx.x;

    // cooperative panel loads (float4, coalesced); D == 64
    for (int i = tid; i < 128 * 16; i += 256) {          // As: 128 rows x 16 float4
        int rr = i >> 4, cc = (i & 15) << 2;
        float4 v = *(const float4*)(Z + (size_t)(row0 + rr) * D + cc);
        *(float4*)&As[rr * ZPAD + cc] = v;
    }
    for (int i = tid; i < 64 * 16; i += 256) {           // Bs: 64 rows x 16 float4
        int rr = i >> 4, cc = (i & 15) << 2;
        float4 v = *(const float4*)(Z + (size_t)(col0 + rr) * D + cc);
        *(float4*)&Bs[rr * ZPAD + cc] = v;
    }
    __syncthreads();

    int wave = tid >> 5, lane = tid & 31;
    int wr = wave & 3;                   // 4 row groups of 32 rows (2 tile-rows)
    int wc = wave >> 2;                  // 2 col groups of 32 cols (2 tile-cols)
    int r = lane & 15, hi = lane >> 4;

    v8f acc[2][2] = {};
    for (int k0 = 0; k0 < D; k0 += 4) {
        v2f a[2], b[2];
        #pragma unroll
        for (int i = 0; i < 2; ++i)
            a[i] = *(const v2f*)&As[(wr * 32 + i * 16 + r) * ZPAD + k0 + 2 * hi];
        #pragma unroll
        for (int j = 0; j < 2; ++j)
            b[j] = *(const v2f*)&Bs[(wc * 32 + j * 16 + r) * ZPAD + k0 + 2 * hi];
        #pragma unroll
        for (int i = 0; i < 2; ++i)
            #pragma unroll
            for (int j = 0; j < 2; ++j)
                acc[i][j] = __builtin_amdgcn_wmma_f32_16x16x4_f32(
                    false, a[i], false, b[j], (short)0, acc[i][j], false, false);
    }

    #pragma unroll
    for (int i = 0; i < 2; ++i) {
        #pragma unroll
        for (int j = 0; j < 2; ++j) {
            #pragma unroll
            for (int v = 0; v < 8; ++v) {
                int row = row0 + wr * 32 + i * 16 + v + 8 * hi;
                int col = col0 + wc * 32 + j * 16 + r;
                float val = acc[i][j][v];
                P[(size_t)row * Nn + col] = 1.0f / (1.0f + expf(-val));
            }
        }
    }
}

// ---------------- misc elementwise / attention kernels ---------------------
__global__ void zero_f32(float* p, size_t n) {
    size_t i = blockIdx.x * (size_t)blockDim.x + threadIdx.x;
    size_t stride = (size_t)gridDim.x * blockDim.x;
    for (; i < n; i += stride) p[i] = 0.0f;
}

__global__ void node_alpha_init(const float* __restrict__ h,
                                const float* __restrict__ a_s,
                                const float* __restrict__ a_d,
                                float* as_, float* ad_, int* mOrd, float* ssum,
                                int Nn, int F)
{
    int i = blockIdx.x * blockDim.x + threadIdx.x;
    if (i >= Nn) return;
    const float* hr = h + (size_t)i * F;
    float s0 = 0.f, s1 = 0.f;
    for (int f = 0; f < F; ++f) { float v = hr[f]; s0 += v * a_s[f]; s1 += v * a_d[f]; }
    as_[i] = s0; ad_[i] = s1;
    mOrd[i] = (int)0x80000000;   // below ord(-inf); every node has a self-loop
    ssum[i] = 0.0f;
}

__global__ void edge_max(const int* __restrict__ ei, const float* __restrict__ as_,
                         const float* __restrict__ ad_, float* __restrict__ eval,
                         int* __restrict__ mOrd, int E, int Nn)
{
    int e = blockIdx.x * blockDim.x + threadIdx.x;
    int ET = E + Nn;
    if (e >= ET) return;
    int s = (e < E) ? ei[e]     : (e - E);
    int d = (e < E) ? ei[E + e] : (e - E);
    float v = as_[s] + ad_[d];
    v = (v > 0.f) ? v : 0.2f * v;            // leaky_relu, slope 0.2
    eval[e] = v;
    atomicMax(mOrd + d, ordf(v));
}

__global__ void edge_expsum(const int* __restrict__ ei, float* __restrict__ eval,
                            const int* __restrict__ mOrd, float* __restrict__ ssum,
                            int E, int Nn)
{
    int e = blockIdx.x * blockDim.x + threadIdx.x;
    int ET = E + Nn;
    if (e >= ET) return;
    int d = (e < E) ? ei[E + e] : (e - E);
    float ex = expf(eval[e] - unordf(mOrd[d]));
    eval[e] = ex;
    atomicAdd(ssum + d, ex);
}

// one block per edge, blockDim.x == F (feature-parallel, coalesced)
__global__ void edge_aggr(const int* __restrict__ ei, const float* __restrict__ eval,
                          const float* __restrict__ ssum, const float* __restrict__ h,
                          float* __restrict__ acc, int E, int Nn, int F)
{
    int e = blockIdx.x;
    int s = (e < E) ? ei[e]     : (e - E);
    int d = (e < E) ? ei[E + e] : (e - E);
    float alpha = eval[e] / ssum[d];
    int f = threadIdx.x;
    atomicAdd(acc + (size_t)d * F + f, alpha * h[(size_t)s * F + f]);
}

__global__ void finalize_elu(float* __restrict__ acc, const float* __restrict__ b,
                             int Nn, int F)
{
    int idx = blockIdx.x * blockDim.x + threadIdx.x;
    if (idx >= Nn * F) return;
    float v = acc[idx] + b[idx % F];
    acc[idx] = (v > 0.f) ? v : (expf(v) - 1.0f);   // ELU(alpha=1)
}

__global__ void finalize_bias(const float* __restrict__ acc, const float* __restrict__ b,
                              float* __restrict__ z, int Nn, int F)
{
    int idx = blockIdx.x * blockDim.x + threadIdx.x;
    if (idx >= Nn * F) return;
    z[idx] = acc[idx] + b[idx % F];
}

// one wave per node: lanes 0..Kc-1 compute q_k, full-wave shuffle reduce
__global__ void q_kernel(const float* __restrict__ z, const float* __restrict__ centers,
                         float* __restrict__ q, int Nn, int D, int Kc)
{
    int i = blockIdx.x;
    int lane = threadIdx.x;
    float qk = 0.0f;
    if (lane < Kc) {
        const float* zr = z + (size_t)i * D;
        const float* cr = centers + (size_t)lane * D;
        float d2 = 0.f;
        for (int dd = 0; dd < D; ++dd) { float t = zr[dd] - cr[dd]; d2 += t * t; }
        qk = 1.0f / (1.0f + d2);
    }
    float tot = qk;                     // wave32 reduction (inactive lanes add 0)
    for (int off = 16; off > 0; off >>= 1) tot += __shfl_xor(tot, off, 32);
    if (lane < Kc) q[(size_t)i * Kc + lane] = qk / tot;
}

// ---------------------------- host orchestration ---------------------------
extern "C" void kernel_launch(void* const* d_in, const int* in_sizes, int n_in,
                              void* d_out, int out_size, void* d_ws, size_t ws_size,
                              hipStream_t stream)
{
    const float* x      = (const float*)d_in[0];
    const int*   ei     = (const int*)  d_in[1];
    const float* W1     = (const float*)d_in[2];
    const float* a_src1 = (const float*)d_in[3];
    const float* a_dst1 = (const float*)d_in[4];
    const float* b1     = (const float*)d_in[5];
    const float* W2     = (const float*)d_in[6];
    const float* a_src2 = (const float*)d_in[7];
    const float* a_dst2 = (const float*)d_in[8];
    const float* b2     = (const float*)d_in[9];
    const float* cent   = (const float*)d_in[10];

    const int H   = in_sizes[3];              // 128
    const int IN  = in_sizes[2] / H;          // 512
    const int Nn  = in_sizes[0] / IN;         // 8192
    const int E   = in_sizes[1] / 2;          // 262144
    const int OUT = in_sizes[7];              // 64
    const int Kc  = in_sizes[10] / OUT;       // 16
    const int ET  = E + Nn;                   // edges + self-loops

    // workspace layout (all f32 except mOrd)
    float* ws   = (float*)d_ws;
    float* h1   = ws;                          // Nn*H  (also reused for h@W2)
    float* acc1 = h1   + (size_t)Nn * H;       // Nn*H  (becomes ELU'd layer-1 out)
    float* acc2 = acc1 + (size_t)Nn * H;       // Nn*OUT
    float* as_  = acc2 + (size_t)Nn * OUT;     // Nn
    float* ad_  = as_  + Nn;                   // Nn
    int*   mOrd = (int*)(ad_ + Nn);            // Nn
    float* ssum = (float*)(mOrd + Nn);         // Nn
    float* eval = ssum + Nn;                   // ET

    float* z_out = (float*)d_out;                       // Nn*OUT
    float* A_out = z_out + (size_t)Nn * OUT;            // Nn*Nn
    float* q_out = A_out + (size_t)Nn * Nn;             // Nn*Kc

    // ---- layer 1: h1 = x @ W1 --------------------------------------------
    {
        int tiles = (Nn / 16) * (H / 16);
        wmma_gemm_f32<<<(tiles + 7) / 8, 256, 0, stream>>>(x, W1, h1, Nn, H, IN, IN, H, H);
    }
    zero_f32<<<1024, 256, 0, stream>>>(acc1, (size_t)Nn * H);
    node_alpha_init<<<(Nn + 255) / 256, 256, 0, stream>>>(h1, a_src1, a_dst1, as_, ad_, mOrd, ssum, Nn, H);
    edge_max   <<<(ET + 255) / 256, 256, 0, stream>>>(ei, as_, ad_, eval, mOrd, E, Nn);
    edge_expsum<<<(ET + 255) / 256, 256, 0, stream>>>(ei, eval, mOrd, ssum, E, Nn);
    edge_aggr  <<<ET, H, 0, stream>>>(ei, eval, ssum, h1, acc1, E, Nn, H);
    finalize_elu<<<(Nn * H + 255) / 256, 256, 0, stream>>>(acc1, b1, Nn, H);

    // ---- layer 2: h1 = acc1 @ W2 (h1 reused) -----------------------------
    {
        int tiles = (Nn / 16) * (OUT / 16);
        wmma_gemm_f32<<<(tiles + 7) / 8, 256, 0, stream>>>(acc1, W2, h1, Nn, OUT, H, H, OUT, OUT);
    }
    zero_f32<<<1024, 256, 0, stream>>>(acc2, (size_t)Nn * OUT);
    node_alpha_init<<<(Nn + 255) / 256, 256, 0, stream>>>(h1, a_src2, a_dst2, as_, ad_, mOrd, ssum, Nn, OUT);
    edge_max   <<<(ET + 255) / 256, 256, 0, stream>>>(ei, as_, ad_, eval, mOrd, E, Nn);
    edge_expsum<<<(ET + 255) / 256, 256, 0, stream>>>(ei, eval, mOrd, ssum, E, Nn);
    edge_aggr  <<<ET, OUT, 0, stream>>>(ei, eval, ssum, h1, acc2, E, Nn, OUT);
    finalize_bias<<<(Nn * OUT + 255) / 256, 256, 0, stream>>>(acc2, b2, z_out, Nn, OUT);

    // ---- A_pred = sigmoid(z z^T), 128x64 macro-tiles via LDS -------------
    {
        int blocks = (Nn >> 7) * (Nn >> 6);
        wmma_zzt_sigmoid<<<blocks, 256, 0, stream>>>(z_out, A_out, Nn, OUT);
    }

    // ---- Student-t soft assignment q -------------------------------------
    q_kernel<<<Nn, 32, 0, stream>>>(z_out, cent, q_out, Nn, OUT, Kc);
}